// ConvCaps_87995289960774
// MI455X (gfx1250) — compile-verified
//
#include <hip/hip_runtime.h>
#include <hip/hip_bf16.h>
#include <math.h>

// ConvCaps EM-routing, fully fused for gfx1250 (MI455X).
// One workgroup (256 threads = 8 wave32) per output position (b,p).
// Votes are recomputed per pass with V_WMMA_F32_16X16X4_F32; everything
// else (pose gather, rr, mu/var, softmax) lives in LDS (~60KB/WG).
//
// WMMA mapping per child capsule k (one wmma per wave per k):
//   A[16x4]  A[M,j] = pose[k, M&3, j]   (rows replicated -> unconditional loads)
//   B[4x16]  cols n = (c,l) pairs owned by this wave: B[j,n] = W[k, c(n), j, l(n)]
//   D[16x16] vgpr r, lane<16: D[r] = vote(c(n), i=r, l(n))   (other rows = dups)

#define S_IN     14
#define SP2      196          // 14*14
#define KW2      9            // kernel^2
#define C_IN     32
#define C_OUT    32
#define BATCH    4
#define P_OUT    144          // 12*12
#define KTOT     288          // 9*32
#define ROUTINGS 3
#define EPSF     1e-7f
#define LN2PI    1.8378770664093453f

typedef __attribute__((ext_vector_type(2))) float v2f;
typedef __attribute__((ext_vector_type(8))) float v8f;

#if defined(__gfx1250__) && __has_builtin(__builtin_amdgcn_wmma_f32_16x16x4_f32)
#define HAVE_WMMA_F32X4 1
#endif

// poseLane: &s_pose[k][(n&3)*4 + j0]   (8B-aligned, lane-strided LDS)
// poseFull: &s_pose[k][0]              (fallback path only)
// wkLane:   W + ((k*32 + c)<<4) + l    (lane's W column base for this k)
__device__ __forceinline__ void votes4(const float* poseFull,
                                       const float* poseLane,
                                       const float* __restrict__ wkLane,
                                       int j0, float Dv[4])
{
#ifdef HAVE_WMMA_F32X4
    v2f A = *(const v2f*)poseLane;       // ds_load_b64, no exec masking
    v2f B;
    B.x = wkLane[ j0      * 4];          // W[k,c,j0  ,l]
    B.y = wkLane[(j0 + 1) * 4];          // W[k,c,j0+1,l]
    v8f D = {};
    D = __builtin_amdgcn_wmma_f32_16x16x4_f32(false, A, false, B,
                                              (short)0, D, false, false);
    Dv[0] = D[0]; Dv[1] = D[1]; Dv[2] = D[2]; Dv[3] = D[3];
#else
    (void)poseLane; (void)j0;
    #pragma unroll
    for (int r = 0; r < 4; ++r) {
        float s = 0.0f;
        #pragma unroll
        for (int j = 0; j < 4; ++j) s += poseFull[r * 4 + j] * wkLane[j * 4];
        Dv[r] = s;
    }
#endif
}

__global__ __launch_bounds__(256)
void convcaps_em_kernel(const float* __restrict__ in_act,
                        const float* __restrict__ in_pose,
                        const float* __restrict__ W,
                        const float* __restrict__ beta_a,
                        const float* __restrict__ beta_v,
                        const int*   __restrict__ cpm,
                        float*       __restrict__ out)
{
    __shared__ float s_pose[KTOT][16];     // gathered child poses
    __shared__ float s_act [KTOT];         // gathered child activations
    __shared__ float s_rr  [KTOT][C_OUT];  // routing weights (also log_p scratch)
    __shared__ float s_mu  [C_OUT][16];
    __shared__ float s_var [C_OUT][16];
    __shared__ float s_rs  [C_OUT];
    __shared__ float s_a   [C_OUT];

    const int tid = threadIdx.x;
    const int bp  = blockIdx.x;            // b*144 + p
    const int b   = bp / P_OUT;
    const int p   = bp - b * P_OUT;

    // ---------------- gather kernel window into LDS ----------------
    for (int k = tid; k < KTOT; k += 256) {
        int kw = k >> 5, ci = k & 31;
        int sp = cpm[p * KW2 + kw];
        int base = (b * SP2 + sp) * C_IN + ci;
        s_act[k] = in_act[base];
        const float4* ps = (const float4*)(in_pose + (size_t)base * 16);
        float4 q0 = ps[0], q1 = ps[1], q2 = ps[2], q3 = ps[3];
        float* dst = s_pose[k];
        dst[ 0]=q0.x; dst[ 1]=q0.y; dst[ 2]=q0.z; dst[ 3]=q0.w;
        dst[ 4]=q1.x; dst[ 5]=q1.y; dst[ 6]=q1.z; dst[ 7]=q1.w;
        dst[ 8]=q2.x; dst[ 9]=q2.y; dst[10]=q2.z; dst[11]=q2.w;
        dst[12]=q3.x; dst[13]=q3.y; dst[14]=q3.z; dst[15]=q3.w;
    }
    for (int idx = tid; idx < KTOT * C_OUT; idx += 256)
        (&s_rr[0][0])[idx] = 1.0f / (float)C_OUT;
    __syncthreads();

    // wave/lane decomposition: wave w owns c in [4w, 4w+4), lane n = (c,l) column
    const int  wv   = tid >> 5;
    const int  L    = tid & 31;
    const int  n    = L & 15;
    const int  hi   = L >> 4;
    const int  cw   = wv * 4 + (n >> 2);
    const int  l    = n & 3;
    const int  j0   = hi ? 2 : 0;
    const int  offA = (n & 3) * 4 + j0;            // lane's loop-invariant A offset
    const bool wr16 = (L < 16);
    const bool wrE  = (L < 16) && (l == 0);

    const float* Wlane0 = W + ((size_t)cw << 4) + l;   // advances by 512 per k

    for (int it = 0; it < ROUTINGS; ++it) {
        // ---------------- M-step ----------------
        float acc1[4] = {0.f, 0.f, 0.f, 0.f};  // sum rp*v   (d = r*4+l)
        float acc2[4] = {0.f, 0.f, 0.f, 0.f};  // sum rp*v*v
        float accr = 0.f;                      // sum rp
        {
            const float* pl = &s_pose[0][0] + offA;
            const float* wk = Wlane0;
            for (int k = 0; k < KTOT; ++k, pl += 16, wk += 512) {
                if ((k & 7) == 0)              // uniform; pull next 8 W slices to L0
                    __builtin_prefetch(wk + 8 * 512, 0, 1);
                float Dv[4];
                votes4(s_pose[k], pl, wk, j0, Dv);
                float rp = s_rr[k][cw] * s_act[k];
                accr += rp;
                #pragma unroll
                for (int r = 0; r < 4; ++r) {
                    float v = Dv[r];
                    acc1[r] += rp * v;
                    acc2[r] += rp * v * v;
                }
            }
        }
        float sum_rp = accr;
        float rs     = sum_rp + EPSF;
        if (wr16) {
            float inv_rs = 1.0f / rs;
            #pragma unroll
            for (int r = 0; r < 4; ++r) {
                float mu  = acc1[r] * inv_rs;
                float num = acc2[r] - 2.0f * mu * acc1[r] + mu * mu * sum_rp;
                float var = fmaxf(num, 0.0f) * inv_rs + EPSF;
                s_mu [cw][r * 4 + l] = mu;
                s_var[cw][r * 4 + l] = var;
            }
            if (l == 0) s_rs[cw] = rs;
        }
        __syncthreads();

        const float inv_temp = 1.0f + (float)it;  // it_min=1, it_max=3, routings=3
        if (tid < C_OUT) {
            int c = tid;
            float slog = 0.0f;
            #pragma unroll
            for (int d = 0; d < 16; ++d) slog += logf(s_var[c][d]);
            float cost = s_rs[c] * (16.0f * beta_v[c] + 0.5f * slog);
            float x = inv_temp * (beta_a[c] - cost);
            s_a[c] = 1.0f / (1.0f + expf(-x));
        }
        __syncthreads();

        // ---------------- E-step ----------------
        if (it < ROUTINGS - 1) {
            float muL[4], ivL[4];
            float lnterm = 0.0f;
            #pragma unroll
            for (int r = 0; r < 4; ++r) {
                float m_ = s_mu [cw][r * 4 + l];
                float v_ = s_var[cw][r * 4 + l];
                muL[r] = m_;
                ivL[r] = 1.0f / v_;
                lnterm += LN2PI + logf(v_);
            }
            float lna = logf(s_a[cw] + EPSF);

            const float* pl = &s_pose[0][0] + offA;
            const float* wk = Wlane0;
            for (int k = 0; k < KTOT; ++k, pl += 16, wk += 512) {
                if ((k & 7) == 0)
                    __builtin_prefetch(wk + 8 * 512, 0, 1);
                float Dv[4];
                votes4(s_pose[k], pl, wk, j0, Dv);
                float q = lnterm;
                #pragma unroll
                for (int r = 0; r < 4; ++r) {
                    float dd = Dv[r] - muL[r];
                    q += dd * dd * ivL[r];
                }
                // sum over the 4 l-lanes sharing this c (stays inside lane quads)
                q += __shfl_xor(q, 1, 32);
                q += __shfl_xor(q, 2, 32);
                if (wrE)
                    s_rr[k][cw] = lna - 0.5f * q;   // ln(a+eps) + log_p
            }
            __syncthreads();

            // softmax over parent capsules c, per child k (in place)
            for (int k = tid; k < KTOT; k += 256) {
                float e[C_OUT];
                float m = -3.0e38f;
                #pragma unroll
                for (int c = 0; c < C_OUT; ++c) {
                    e[c] = s_rr[k][c];
                    m = fmaxf(m, e[c]);
                }
                float ssum = 0.0f;
                #pragma unroll
                for (int c = 0; c < C_OUT; ++c) {
                    e[c] = expf(e[c] - m);
                    ssum += e[c];
                }
                float inv = 1.0f / ssum;
                #pragma unroll
                for (int c = 0; c < C_OUT; ++c)
                    s_rr[k][c] = e[c] * inv;
            }
            __syncthreads();
        }
    }

    // ---------------- outputs: [out_act flat | out_pose flat] ----------------
    if (tid < C_OUT)
        out[(size_t)bp * C_OUT + tid] = s_a[tid];
    const size_t pose_base = (size_t)BATCH * P_OUT * C_OUT;   // 18432
    for (int idx = tid; idx < C_OUT * 16; idx += 256)
        out[pose_base + ((size_t)bp << 9) + idx] = (&s_mu[0][0])[idx];
}

extern "C" void kernel_launch(void* const* d_in, const int* in_sizes, int n_in,
                              void* d_out, int out_size, void* d_ws, size_t ws_size,
                              hipStream_t stream) {
    (void)in_sizes; (void)n_in; (void)out_size; (void)d_ws; (void)ws_size;
    const float* in_act  = (const float*)d_in[0];
    const float* in_pose = (const float*)d_in[1];
    const float* W       = (const float*)d_in[2];
    const float* beta_a  = (const float*)d_in[3];
    const float* beta_v  = (const float*)d_in[4];
    const int*   cpm     = (const int*)  d_in[5];
    float* out = (float*)d_out;

    convcaps_em_kernel<<<dim3(BATCH * P_OUT), dim3(256), 0, stream>>>(
        in_act, in_pose, W, beta_a, beta_v, cpm, out);
}